// MinimalLightningIndexer_57604101374285
// MI455X (gfx1250) — compile-verified
//
#include <hip/hip_runtime.h>

typedef __attribute__((ext_vector_type(2))) float v2f;
typedef __attribute__((ext_vector_type(4))) float v4f;
typedef __attribute__((ext_vector_type(8))) float v8f;

#define D_MODEL 2048
#define IDX_DIM 16
#define SEQ     4096
#define NBATCH  4
#define NROWS   (NBATCH * SEQ)   // 16384 flattened (b,s) rows

// ---------------------------------------------------------------------------
// Phase 1: q = x@Wq, k = x@Wk (V_WMMA_F32_16X16X4_F32), w = x@Ww (VALU).
// One block per 16-row M-tile; 4 waves split the D=2048 reduction; LDS
// ds_add_f32 combines partial accumulators. x is streamed with non-temporal
// loads (read-once; short-range line reuse is caught by L0) so the q/k/w
// results stay resident in the 192 MB L2 for phase 2.
// ---------------------------------------------------------------------------
__global__ __launch_bounds__(128) void proj_kernel(
    const float* __restrict__ x,
    const float* __restrict__ Wq,
    const float* __restrict__ Wk,
    const float* __restrict__ Ww,
    float* __restrict__ qo, float* __restrict__ ko, float* __restrict__ wo)
{
    __shared__ float lds[528];             // 256 q + 256 k + 16 w
    const int tid  = threadIdx.x;
    const int lane = tid & 31;
    const int wv   = tid >> 5;             // wave 0..3
    const int c    = lane & 15;            // matrix row (A) / col (B)
    const int h    = lane >> 4;            // half-wave select
    const int m0   = blockIdx.x * 16;      // flattened row base

    for (int e = tid; e < 528; e += 128) lds[e] = 0.0f;
    __syncthreads();

    const float* xrow = x + (size_t)(m0 + c) * D_MODEL;

    v8f accq = {};
    v8f acck = {};
    float wpart = 0.0f;

    const int kbeg = wv * (D_MODEL / 4);   // 512 floats per wave
    const int kend = kbeg + (D_MODEL / 4);
    #pragma unroll 4
    for (int k0 = kbeg; k0 < kend; k0 += 4) {
        const int kk = k0 + 2 * h;
        // A tile (x): lane holds {x[m0+c][kk], x[m0+c][kk+1]}  (streaming)
        v2f a = __builtin_nontemporal_load((const v2f*)(xrow + kk));
        // B tiles (Wq/Wk columns): lane holds {W[kk][c], W[kk+1][c]}
        v2f bq, bk;
        bq.x = Wq[(size_t)kk * IDX_DIM + c];
        bq.y = Wq[(size_t)(kk + 1) * IDX_DIM + c];
        bk.x = Wk[(size_t)kk * IDX_DIM + c];
        bk.y = Wk[(size_t)(kk + 1) * IDX_DIM + c];
        v2f ww = *(const v2f*)(Ww + kk);

        accq = __builtin_amdgcn_wmma_f32_16x16x4_f32(
            false, a, false, bq, (short)0, accq, false, false);
        acck = __builtin_amdgcn_wmma_f32_16x16x4_f32(
            false, a, false, bk, (short)0, acck, false, false);
        wpart = __builtin_fmaf(a.x, ww.x, wpart);
        wpart = __builtin_fmaf(a.y, ww.y, wpart);
    }

    // Combine the 4 waves' partials in LDS (ds_add_f32 atomics).
    #pragma unroll
    for (int r = 0; r < 8; ++r) {
        const int idx = (r + 8 * h) * 16 + c;   // C/D layout: (m=r+8h, n=c)
        atomicAdd(&lds[idx],       accq[r]);
        atomicAdd(&lds[256 + idx], acck[r]);
    }
    atomicAdd(&lds[512 + c], wpart);            // row c partial of w
    __syncthreads();

    // Coalesced stores: tile is 16 consecutive rows of [rows][16] arrays.
    for (int e = tid; e < 528; e += 128) {
        const float v = lds[e];
        if (e < 256)      qo[(size_t)m0 * IDX_DIM + e]         = v;
        else if (e < 512) ko[(size_t)m0 * IDX_DIM + (e - 256)] = v;
        else              wo[m0 + (e - 512)]                   = v;
    }
}

// ---------------------------------------------------------------------------
// Phase 2: out[b][t][s] = relu(q[b,t,:] . k[b,s,:]) * w[b,t]
// One block per (batch, 16-row t-tile); each wave keeps its Q tile + gates in
// registers and streams 32 s-tiles: 4 chained f32 WMMAs -> relu*gate -> store.
// Output (256 MB, write-once) uses non-temporal stores so it does not evict
// the L2-resident q/k/w buffers that every block re-reads.
// ---------------------------------------------------------------------------
__global__ __launch_bounds__(256) void dots_kernel(
    const float* __restrict__ q,
    const float* __restrict__ k,
    const float* __restrict__ w,
    float* __restrict__ out)
{
    const int tid  = threadIdx.x;
    const int lane = tid & 31;
    const int wv   = tid >> 5;             // wave 0..7
    const int c    = lane & 15;
    const int h    = lane >> 4;
    const int bt   = blockIdx.x;           // 0..1023
    const int b    = bt >> 8;
    const int t0   = (bt & 255) * 16;
    const int row0 = b * SEQ + t0;         // flattened q row base

    // A-matrix (queries), 4 K-steps of 4: lane holds {Q[t0+c][k0+2h], +1}
    const float* qrow = q + (size_t)(row0 + c) * IDX_DIM + 2 * h;
    const v2f a0 = *(const v2f*)(qrow + 0);
    const v2f a1 = *(const v2f*)(qrow + 4);
    const v2f a2 = *(const v2f*)(qrow + 8);
    const v2f a3 = *(const v2f*)(qrow + 12);

    // Gate values for the 8 output rows this lane owns (m = r + 8h).
    const v4f wlo = *(const v4f*)(w + row0 + 8 * h);
    const v4f whi = *(const v4f*)(w + row0 + 8 * h + 4);

    float* outb = out + (size_t)b * SEQ * SEQ + (size_t)t0 * SEQ;
    const float* kb = k + (size_t)b * SEQ * IDX_DIM;

    #pragma unroll 2
    for (int j = 0; j < 32; ++j) {
        const int s0 = (wv + 8 * j) * 16;
        // B-matrix = K^T tile: identical lane pattern as A but over s-rows.
        const float* krow = kb + (size_t)(s0 + c) * IDX_DIM + 2 * h;
        const v2f b0 = *(const v2f*)(krow + 0);
        const v2f b1 = *(const v2f*)(krow + 4);
        const v2f b2 = *(const v2f*)(krow + 8);
        const v2f b3 = *(const v2f*)(krow + 12);

        v8f acc = {};
        acc = __builtin_amdgcn_wmma_f32_16x16x4_f32(
            false, a0, false, b0, (short)0, acc, false, false);
        acc = __builtin_amdgcn_wmma_f32_16x16x4_f32(
            false, a1, false, b1, (short)0, acc, false, false);
        acc = __builtin_amdgcn_wmma_f32_16x16x4_f32(
            false, a2, false, b2, (short)0, acc, false, false);
        acc = __builtin_amdgcn_wmma_f32_16x16x4_f32(
            false, a3, false, b3, (short)0, acc, false, false);

        #pragma unroll
        for (int r = 0; r < 8; ++r) {
            const float g = (r < 4) ? wlo[r] : whi[r - 4];
            float v = acc[r];
            v = v > 0.0f ? v : 0.0f;
            __builtin_nontemporal_store(
                v * g, &outb[(size_t)(r + 8 * h) * SEQ + s0 + c]);
        }
    }
}

// ---------------------------------------------------------------------------
extern "C" void kernel_launch(void* const* d_in, const int* in_sizes, int n_in,
                              void* d_out, int out_size, void* d_ws, size_t ws_size,
                              hipStream_t stream)
{
    const float* x  = (const float*)d_in[0];
    const float* Wq = (const float*)d_in[1];
    const float* Wk = (const float*)d_in[2];
    const float* Ww = (const float*)d_in[3];

    float* qbuf = (float*)d_ws;                          // 16384*16 f32 = 1 MB
    float* kbuf = qbuf + (size_t)NROWS * IDX_DIM;        // 1 MB
    float* wbuf = kbuf + (size_t)NROWS * IDX_DIM;        // 64 KB

    proj_kernel<<<NROWS / 16, 128, 0, stream>>>(x, Wq, Wk, Ww, qbuf, kbuf, wbuf);
    dots_kernel<<<NBATCH * (SEQ / 16), 256, 0, stream>>>(qbuf, kbuf, wbuf,
                                                         (float*)d_out);
}